// Optimizee_72722386256465
// MI455X (gfx1250) — compile-verified
//
#include <hip/hip_runtime.h>
#include <math.h>

// Problem geometry
#define SYS_LEN   16384
#define FFT_LEN   32768          // 2*SYS_LEN = 16*16*16*8
#define N_BLOCKS  512
#define NSPLIT    8              // b-dimension split for the streaming reduce
#define BPS       (N_BLOCKS / NSPLIT)
#define TROWS     4              // rows (b-values) per TDM tile

typedef float v2f __attribute__((ext_vector_type(2)));
typedef float v4f __attribute__((ext_vector_type(4)));
typedef float v8f __attribute__((ext_vector_type(8)));
typedef unsigned int v4u __attribute__((ext_vector_type(4)));
typedef int v4i __attribute__((ext_vector_type(4)));
typedef int v8i __attribute__((ext_vector_type(8)));

// ---------------------------------------------------------------------------
// 1) ClipSigNonlin, writes complex (imag=0) into FFT ping buffer
// ---------------------------------------------------------------------------
__global__ void k_nonlin(const float* __restrict__ u,
                         const float* __restrict__ xmax,
                         const float* __restrict__ gamma,
                         const float* __restrict__ a1,
                         const float* __restrict__ a2,
                         float* __restrict__ out /* interleaved ri */) {
    int i = blockIdx.x * blockDim.x + threadIdx.x;   // FFT_LEN threads
    float xm = fabsf(xmax[0]);
    float uu = u[i];
    float c  = xm * uu * rsqrtf(xm * xm + uu * uu);
    float si = -(a1[0] * c + a2[0] * c * c);
    float v  = gamma[0] * (2.0f / (1.0f + __expf(si)) - 1.0f);
    out[2 * i]     = v;
    out[2 * i + 1] = 0.0f;
}

// ---------------------------------------------------------------------------
// 2) Radix-16 Stockham DIF stage via V_WMMA_F32_16X16X4_F32.
//    Y(16 x cols) = D16(16x16) * X(16 x cols), then twiddle W_n^{p*j}.
//    One wave owns one 16x16 column tile; K=16 accumulated as 4 chained
//    K=4 WMMAs; complex product = 4 real matmuls (C,S,-S twiddle matrices).
//    cols = (n/16)*s = 2048 for every stage -> 128 waves.
// ---------------------------------------------------------------------------
__global__ void k_fft_r16_wmma(const float* __restrict__ xin,
                               float* __restrict__ yout,
                               int n, int s, int ls /* log2(s) */) {
    const int lane = threadIdx.x & 31;
    const int wave = blockIdx.x * (blockDim.x >> 5) + (threadIdx.x >> 5);
    const int m    = n >> 4;

    const int hi  = lane >> 4;        // lane half selects K/M sub-range
    const int lm  = lane & 15;        // A: row M;  B/D: column N
    const int col = wave * 16 + lm;   // global column of this lane's B/D slot
    const int q   = col & (s - 1);
    const int p   = col >> ls;

    const float2* __restrict__ xc = (const float2*)xin;
    float2*       __restrict__ yc = (float2*)yout;

    v8f yr = {};   // Re accumulator (16x16 tile)
    v8f yi = {};   // Im accumulator

#pragma unroll
    for (int kc = 0; kc < 4; ++kc) {
        // ---- A operand: D16 columns K = kc*4 + {2*hi, 2*hi+1}, row M = lm.
        // D16[m][k] = exp(-i*2pi*m*k/16) = cos(th) - i*sin(th), th = pi*m*k/8
        const int k0 = kc * 4 + 2 * hi;
        float s0, c0, s1, c1;
        sincospif((float)((lm * k0) & 15) * 0.125f, &s0, &c0);
        sincospif((float)((lm * (k0 + 1)) & 15) * 0.125f, &s1, &c1);
        v2f aC  = { c0,  c1};
        v2f aS  = { s0,  s1};   //  +sin  (since Di = -sin)
        v2f aSn = {-s0, -s1};

        // ---- B operand: X rows r = kc*4 + {hi, hi+2}, column = col.
        const int r0 = kc * 4 + hi;
        float2 b0 = xc[q + s * (p + m * r0)];
        float2 b1 = xc[q + s * (p + m * (r0 + 2))];
        v2f bXr = {b0.x, b1.x};
        v2f bXi = {b0.y, b1.y};

        // Yr = C*Xr + S*Xi ; Yi = C*Xi - S*Xr
        yr = __builtin_amdgcn_wmma_f32_16x16x4_f32(false, aC,  false, bXr,
                                                   (short)0, yr, false, false);
        yr = __builtin_amdgcn_wmma_f32_16x16x4_f32(false, aS,  false, bXi,
                                                   (short)0, yr, false, false);
        yi = __builtin_amdgcn_wmma_f32_16x16x4_f32(false, aC,  false, bXi,
                                                   (short)0, yi, false, false);
        yi = __builtin_amdgcn_wmma_f32_16x16x4_f32(false, aSn, false, bXr,
                                                   (short)0, yi, false, false);
    }

    // ---- twiddle W_n^{p*j} and scatter: D[M=v+8*hi][N=lm], out row j = M.
    const float ninv = 1.0f / (float)n;
#pragma unroll
    for (int v = 0; v < 8; ++v) {
        const int j = v + 8 * hi;
        float tws, twc;
        sincospif(-2.0f * (float)((p * j) & (n - 1)) * ninv, &tws, &twc);
        float re = yr[v] * twc - yi[v] * tws;
        float im = yr[v] * tws + yi[v] * twc;
        yc[q + s * (16 * p + j)] = make_float2(re, im);
    }
}

// ---------------------------------------------------------------------------
// 3) Final radix-8 stage (n=8, s=4096, p=0 -> unit twiddles). One thread per
//    column q does a direct 8-point DFT (compute is negligible).
// ---------------------------------------------------------------------------
__global__ void k_fft_r8(const float* __restrict__ xin,
                         float* __restrict__ yout) {
    const int q = blockIdx.x * blockDim.x + threadIdx.x;  // 4096 threads
    const int s = FFT_LEN / 8;
    const float2* __restrict__ xc = (const float2*)xin;
    float2*       __restrict__ yc = (float2*)yout;
    float2 a[8];
#pragma unroll
    for (int r = 0; r < 8; ++r) a[r] = xc[q + s * r];
#pragma unroll
    for (int j = 0; j < 8; ++j) {
        float accr = 0.0f, acci = 0.0f;
#pragma unroll
        for (int r = 0; r < 8; ++r) {
            float sn, cs;
            sincospif(-(float)((j * r) & 7) * 0.25f, &sn, &cs);
            accr += a[r].x * cs - a[r].y * sn;
            acci += a[r].x * sn + a[r].y * cs;
        }
        yc[q + s * j] = make_float2(accr, acci);
    }
}

// ---------------------------------------------------------------------------
// 4) TDM helper: issue a 2D tensor_load_to_lds of `rows` rows x 256B from a
//    row-major [*, 65536 f32] tensor slice into LDS at byte offset ldsOff.
//    D# group0/group1 bitfields per CDNA5 ISA ch.8 (data_size=4B,
//    tile_dim0=tensor_dim0=64 elems, tensor_dim0_stride=65536 elems).
//    Toolchain uses the 6-arg builtin form (extra int32x8 group + cpol).
// ---------------------------------------------------------------------------
__device__ __forceinline__ void tdm_load_2d(unsigned int ldsOff,
                                            const void* gptr, int rows) {
    unsigned long long ga = (unsigned long long)(uintptr_t)gptr;
    v4u g0;
    g0.x = 1u;                                   // count=1, user descriptor
    g0.y = ldsOff;                               // lds_addr [63:32]
    g0.z = (unsigned int)ga;                     // global_addr [95:64]
    g0.w = ((unsigned int)(ga >> 32) & 0x01FFFFFFu) | (2u << 30); // addr|type=2
    v8i g1;
    g1[0] = 0x00020000;                          // wg_mask=0, data_size=2 (4B)
    g1[1] = (int)((64u & 0xFFFFu) << 16);        // tensor_dim0 lo16 @bit48
    g1[2] = (int)(((unsigned)rows & 0xFFFFu) << 16); // tdim0 hi=0 | tensor_dim1 lo
    g1[3] = (int)(64u << 16);                    // tdim1 hi=0 | tile_dim0=64
    g1[4] = (int)((unsigned)rows & 0xFFFFu);     // tile_dim1=rows | tile_dim2=0
    g1[5] = (int)(2u * FFT_LEN);                 // tensor_dim0_stride = 65536
    g1[6] = 0;                                   // stride0 hi | stride1 lo
    g1[7] = 0;                                   // stride1 hi
    v4i gz4 = {0, 0, 0, 0};                      // groups 2/3: 2D tensor
    v8i gz8 = {0, 0, 0, 0, 0, 0, 0, 0};          // trailing group (unused)
    __builtin_amdgcn_tensor_load_to_lds(g0, g1, gz4, gz4, gz8, 0);
}

// ---------------------------------------------------------------------------
// 5) Streaming multiply-reduce via TDM double-buffering, fused state roll.
//    Y = W[0]*U + sum_{bb=0..510} W[bb+1]*buf[bb]   (U-term done in combine)
//    split owns bb in [split*BPS, split*BPS+nrows); also writes
//    cur_buffer[bb+1] = buf[bb] from the LDS tile (the roll).
//    Each wave: disjoint LDS region, its own TENSORcnt pipeline -> no
//    barriers. 2 streams (W,buf) x double buffer x TROWS x 256B = 4KB/wave.
// ---------------------------------------------------------------------------
__global__ void k_reduce_tdm(const float* __restrict__ W,
                             const float* __restrict__ buf,
                             float* __restrict__ outbuf,   // d_out + SYS_LEN
                             float* __restrict__ partial) {
    __shared__ v2f ldsW[2][8][TROWS][32];   // [dbuf][wave][row][laneF2] 16KB
    __shared__ v2f ldsV[2][8][TROWS][32];   // 16KB
    const int lane  = threadIdx.x & 31;
    const int wv    = threadIdx.x >> 5;
    const int split = blockIdx.y;
    const int colF2 = (blockIdx.x * 8 + wv) * 32 + lane;  // complex column
    const int colF  = (blockIdx.x * 8 + wv) * 64;         // float column base
    const int base  = split * BPS;                        // bb base
    const int nrows = (split == NSPLIT - 1) ? (BPS - 1) : BPS;
    const int ntile = (nrows + TROWS - 1) / TROWS;

    // prologue: tile 0 into double-buffer slot 0
    {
        int rt = nrows < TROWS ? nrows : TROWS;
        tdm_load_2d((unsigned int)(uintptr_t)&ldsV[0][wv][0][0],
                    buf + (size_t)base * (2 * FFT_LEN) + colF, rt);
        tdm_load_2d((unsigned int)(uintptr_t)&ldsW[0][wv][0][0],
                    W + (size_t)(base + 1) * (2 * FFT_LEN) + colF, rt);
    }

    v2f acc = {0.0f, 0.0f};
    for (int t = 0; t < ntile; ++t) {
        const int db = t & 1;
        if (t + 1 < ntile) {                 // issue next tile, then wait for
            const int nb = (t + 1) & 1;      // the oldest pair (in-order TDM)
            int rleft = nrows - (t + 1) * TROWS;
            int rt = rleft < TROWS ? rleft : TROWS;
            tdm_load_2d((unsigned int)(uintptr_t)&ldsV[nb][wv][0][0],
                        buf + (size_t)(base + (t + 1) * TROWS) * (2 * FFT_LEN) + colF, rt);
            tdm_load_2d((unsigned int)(uintptr_t)&ldsW[nb][wv][0][0],
                        W + (size_t)(base + (t + 1) * TROWS + 1) * (2 * FFT_LEN) + colF, rt);
            __builtin_amdgcn_s_wait_tensorcnt((short)2);
        } else {
            __builtin_amdgcn_s_wait_tensorcnt((short)0);
        }
        int rows_t = nrows - t * TROWS;
        if (rows_t > TROWS) rows_t = TROWS;
        for (int r = 0; r < rows_t; ++r) {
            v2f w2 = ldsW[db][wv][r][lane];
            v2f vv = ldsV[db][wv][r][lane];
            acc.x += w2.x * vv.x - w2.y * vv.y;
            acc.y += w2.x * vv.y + w2.y * vv.x;
            const int bb = base + t * TROWS + r;          // roll: out[bb+1]
            __builtin_nontemporal_store(
                vv, (v2f*)outbuf + (size_t)(bb + 1) * FFT_LEN + colF2);
        }
    }
    ((v2f*)partial)[(size_t)split * FFT_LEN + colF2] = acc;
}

// ---------------------------------------------------------------------------
// 6) Combine partials + W[0]*U term, write cur_buffer[0]=U, conjugate
//    (ifft(Y) = conj(fft(conj(Y)))/N)
// ---------------------------------------------------------------------------
__global__ void k_combine(const float* __restrict__ partial,
                          const float* __restrict__ W,
                          const float* __restrict__ U,
                          float* __restrict__ outbuf,    // d_out + SYS_LEN
                          float* __restrict__ Yc) {
    const int tid = blockIdx.x * blockDim.x + threadIdx.x;   // 16384
    const int s4  = FFT_LEN / 2;
    const v4f* P4 = (const v4f*)partial;
    v4f acc = {0.0f, 0.0f, 0.0f, 0.0f};
#pragma unroll
    for (int s = 0; s < NSPLIT; ++s) {
        acc += P4[(size_t)s * s4 + tid];
    }
    v4f uu = ((const v4f*)U)[tid];
    v4f w  = __builtin_nontemporal_load((const v4f*)W + tid);  // W row 0
    acc.x += w.x * uu.x - w.y * uu.y;  acc.y += w.x * uu.y + w.y * uu.x;
    acc.z += w.z * uu.z - w.w * uu.w;  acc.w += w.z * uu.w + w.w * uu.z;
    __builtin_nontemporal_store(uu, (v4f*)outbuf + tid);       // cur_buffer[0]
    acc.y = -acc.y;  acc.w = -acc.w;                           // conj
    ((v4f*)Yc)[tid] = acc;
}

// 7) y[t] = Re(conj(R))[SYS_LEN + t]/N = Re(R)[SYS_LEN + t]/N
__global__ void k_final(const float* __restrict__ R, float* __restrict__ y) {
    const int i = blockIdx.x * blockDim.x + threadIdx.x;     // SYS_LEN
    y[i] = R[2 * (SYS_LEN + i)] * (1.0f / (float)FFT_LEN);
}

// ---------------------------------------------------------------------------
extern "C" void kernel_launch(void* const* d_in, const int* in_sizes, int n_in,
                              void* d_out, int out_size, void* d_ws, size_t ws_size,
                              hipStream_t stream) {
    const float* u     = (const float*)d_in[0];
    const float* w     = (const float*)d_in[1];
    const float* buf   = (const float*)d_in[2];
    const float* xmax  = (const float*)d_in[3];
    const float* gamma = (const float*)d_in[4];
    const float* a1    = (const float*)d_in[5];
    const float* a2    = (const float*)d_in[6];
    float* out = (float*)d_out;                 // [SYS_LEN] y ++ [512][2S][2]
    float* ws  = (float*)d_ws;

    float* bufA    = ws;                         // 65536 f
    float* bufB    = ws + 2 * FFT_LEN;           // 65536 f
    float* partial = ws + 4 * FFT_LEN;           // NSPLIT * 65536 f

    // nonlinearity -> complex input in bufA
    k_nonlin<<<FFT_LEN / 256, 256, 0, stream>>>(u, xmax, gamma, a1, a2, bufA);

    // forward FFT: 16,16,16,8 Stockham (ping-pong A->B->A->B->A), U in bufA
    k_fft_r16_wmma<<<16, 256, 0, stream>>>(bufA, bufB, 32768, 1,   0);
    k_fft_r16_wmma<<<16, 256, 0, stream>>>(bufB, bufA, 2048,  16,  4);
    k_fft_r16_wmma<<<16, 256, 0, stream>>>(bufA, bufB, 128,   256, 8);
    k_fft_r8<<<16, 256, 0, stream>>>(bufB, bufA);

    // TDM double-buffered streaming reduce + state roll; then combine
    dim3 rg(128, NSPLIT);
    k_reduce_tdm<<<rg, 256, 0, stream>>>(w, buf, out + SYS_LEN, partial);
    k_combine<<<64, 256, 0, stream>>>(partial, w, bufA, out + SYS_LEN, bufB);

    // inverse FFT via forward-on-conjugate (B->A->B->A->B), result in bufB
    k_fft_r16_wmma<<<16, 256, 0, stream>>>(bufB, bufA, 32768, 1,   0);
    k_fft_r16_wmma<<<16, 256, 0, stream>>>(bufA, bufB, 2048,  16,  4);
    k_fft_r16_wmma<<<16, 256, 0, stream>>>(bufB, bufA, 128,   256, 8);
    k_fft_r8<<<16, 256, 0, stream>>>(bufA, bufB);

    k_final<<<SYS_LEN / 256, 256, 0, stream>>>(bufB, out);
}